// ConceptEmbedModel_18262200942819
// MI455X (gfx1250) — compile-verified
//
#include <hip/hip_runtime.h>
#include <math.h>

typedef __attribute__((ext_vector_type(2))) float v2f;
typedef __attribute__((ext_vector_type(8))) float v8f;

#define B_  64
#define L_  2048
#define F_  100
#define D_  1024
#define NC  50000

// ---------------------------------------------------------------------------
// Kernel 1: out1 = elu(x @ conv_w^T + conv_b), pooled = max over L.
// Grid: (8 d-tiles of 128, 64 b). Block: 256 threads = 8 waves.
// Each wave owns 16 d-columns; loops over L in 16-row WMMA tiles.
// sched_barrier(0) pins all 25 A-tile loads ahead of the WMMA chain so the
// wave keeps 25 loads in flight; two accumulators halve the WMMA RAW chain.
// ---------------------------------------------------------------------------
__global__ __launch_bounds__(256) void k1_conv_elu_maxpool(
    const float* __restrict__ x, const float* __restrict__ conv_w,
    const float* __restrict__ conv_b, float* __restrict__ pooled)
{
  const int wave = threadIdx.x >> 5;
  const int lane = threadIdx.x & 31;
  const int n    = lane & 15;        // column within wave tile / row M for A
  const int koff = (lane >> 4) << 1; // K offset 0 or 2
  const int d0   = blockIdx.x * 128 + wave * 16;
  const int b    = blockIdx.y;

  // Preload B fragments (conv_w tile, kept in registers across the L loop):
  // B[k][n] = conv_w[(d0+n)*F + 4k + koff .. +1]
  v2f bfrag[25];
  const float* wp = conv_w + (size_t)(d0 + n) * F_;
#pragma unroll
  for (int k = 0; k < 25; ++k)
    bfrag[k] = *(const v2f*)(wp + 4 * k + koff);

  const float bias = conv_b[d0 + n];

  v8f mx;
#pragma unroll
  for (int r = 0; r < 8; ++r) mx[r] = -3.4e38f;

  const float* xb = x + (size_t)b * L_ * F_;
  for (int l0 = 0; l0 < L_; l0 += 16) {
    const float* xp = xb + (size_t)(l0 + n) * F_;
    // preload the full A tile: 25 b64 loads, all outstanding simultaneously
    v2f af[25];
#pragma unroll
    for (int k = 0; k < 25; ++k)
      af[k] = *(const v2f*)(xp + 4 * k + koff);

    // keep the loads above the WMMA chain (max memory-level parallelism)
    __builtin_amdgcn_sched_barrier(0);

    v8f acc0 = {}, acc1 = {};
#pragma unroll
    for (int k = 0; k < 24; k += 2) {
      acc0 = __builtin_amdgcn_wmma_f32_16x16x4_f32(
          false, af[k], false, bfrag[k], (short)0, acc0, false, false);
      acc1 = __builtin_amdgcn_wmma_f32_16x16x4_f32(
          false, af[k + 1], false, bfrag[k + 1], (short)0, acc1, false, false);
    }
    acc0 = __builtin_amdgcn_wmma_f32_16x16x4_f32(
        false, af[24], false, bfrag[24], (short)0, acc0, false, false);

    // bias + ELU + running max over the 16 rows of this tile
#pragma unroll
    for (int r = 0; r < 8; ++r) {
      float z = acc0[r] + acc1[r] + bias;
      float e = z > 0.0f ? z : expm1f(z);
      mx[r] = fmaxf(mx[r], e);
    }
  }
  float m = mx[0];
#pragma unroll
  for (int r = 1; r < 8; ++r) m = fmaxf(m, mx[r]);
  m = fmaxf(m, __shfl_xor(m, 16, 32)); // combine rows 0-7 with 8-15
  if (lane < 16) pooled[(size_t)b * D_ + d0 + n] = m;
}

// ---------------------------------------------------------------------------
// Kernel 2: out2 = relu(pooled @ lin_w^T + lin_b); xn = row-normalize(out2).
// Grid: 64 (one block per row b). Block: 256.
// ---------------------------------------------------------------------------
__global__ __launch_bounds__(256) void k2_fc_relu_norm(
    const float* __restrict__ pooled, const float* __restrict__ lin_w,
    const float* __restrict__ lin_b, float* __restrict__ xn)
{
  __shared__ float sp[D_];
  __shared__ float srow[D_];
  __shared__ float red[256];
  const int tid = threadIdx.x;
  const int b   = blockIdx.x;

  for (int i = tid; i < D_; i += 256) sp[i] = pooled[(size_t)b * D_ + i];
  __syncthreads();

  float ssq = 0.f;
#pragma unroll
  for (int r = 0; r < 4; ++r) {
    const int j = tid + 256 * r;
    const float4* wr = (const float4*)(lin_w + (size_t)j * D_);
    float acc = lin_b[j];
    for (int d4 = 0; d4 < D_ / 4; ++d4) {
      float4 w = wr[d4];
      acc += w.x * sp[4 * d4] + w.y * sp[4 * d4 + 1] +
             w.z * sp[4 * d4 + 2] + w.w * sp[4 * d4 + 3];
    }
    acc = fmaxf(acc, 0.f);
    srow[j] = acc;
    ssq += acc * acc;
  }
  red[tid] = ssq;
  __syncthreads();
  for (int s = 128; s > 0; s >>= 1) {
    if (tid < s) red[tid] += red[tid + s];
    __syncthreads();
  }
  const float inv = 1.0f / fmaxf(sqrtf(red[0]), 1e-12f);
  for (int i = tid; i < D_; i += 256)
    xn[(size_t)b * D_ + i] = srow[i] * inv;
}

// ---------------------------------------------------------------------------
// Kernel 3: logits = xn @ H^T via WMMA f32 16x16x4.
// Grid: 3125 (N-tiles of 16 concepts). Block: 128 = 4 waves (M-tiles of 16 b).
// All 4 waves share the same 16-concept H tile -> H streams once (~205 MB).
// H loads are non-temporal (no cross-block reuse; keep L2 for hot data).
// Each 8-step chunk issues its 16 loads before its WMMAs (sched_barrier).
// ---------------------------------------------------------------------------
__global__ __launch_bounds__(128) void k3_logits_wmma(
    const float* __restrict__ xn, const float* __restrict__ H,
    float* __restrict__ out)
{
  const int wave = threadIdx.x >> 5;
  const int lane = threadIdx.x & 31;
  const int n    = lane & 15;
  const int half = lane >> 4;
  const int koff = half << 1;
  const int m0   = wave * 16;
  const int c0   = blockIdx.x * 16;

  const float* ap = xn + (size_t)(m0 + n) * D_;
  const float* bp = H  + (size_t)(c0 + n) * D_;
  v8f acc0 = {}, acc1 = {};
  for (int kc = 0; kc < D_ / 4; kc += 8) {
    v2f af[8], bf[8];
#pragma unroll
    for (int u = 0; u < 8; ++u) {
      af[u] = *(const v2f*)(ap + 4 * (kc + u) + koff);
      bf[u] = __builtin_nontemporal_load((const v2f*)(bp + 4 * (kc + u) + koff));
    }
    __builtin_amdgcn_sched_barrier(0);
#pragma unroll
    for (int u = 0; u < 8; u += 2) {
      acc0 = __builtin_amdgcn_wmma_f32_16x16x4_f32(
          false, af[u], false, bf[u], (short)0, acc0, false, false);
      acc1 = __builtin_amdgcn_wmma_f32_16x16x4_f32(
          false, af[u + 1], false, bf[u + 1], (short)0, acc1, false, false);
    }
  }
#pragma unroll
  for (int r = 0; r < 8; ++r) {
    const int row = m0 + half * 8 + r;
    out[(size_t)row * NC + c0 + n] = acc0[r] + acc1[r];
  }
}

// ---------------------------------------------------------------------------
// Kernel 4: in-place row softmax over 50000 concepts. Grid: 64, block: 512.
// Deterministic LDS tree reductions (no float atomics).
// ---------------------------------------------------------------------------
__global__ __launch_bounds__(512) void k4_softmax(float* __restrict__ out)
{
  __shared__ float red[512];
  const int tid = threadIdx.x;
  float* row = out + (size_t)blockIdx.x * NC;

  float mx = -3.4e38f;
  for (int i = tid; i < NC; i += 512) mx = fmaxf(mx, row[i]);
  red[tid] = mx;
  __syncthreads();
  for (int s = 256; s > 0; s >>= 1) {
    if (tid < s) red[tid] = fmaxf(red[tid], red[tid + s]);
    __syncthreads();
  }
  const float rmax = red[0];
  __syncthreads();

  float sum = 0.f;
  for (int i = tid; i < NC; i += 512) sum += expf(row[i] - rmax);
  red[tid] = sum;
  __syncthreads();
  for (int s = 256; s > 0; s >>= 1) {
    if (tid < s) red[tid] += red[tid + s];
    __syncthreads();
  }
  const float inv = 1.0f / red[0];
  for (int i = tid; i < NC; i += 512) row[i] = expf(row[i] - rmax) * inv;
}

// ---------------------------------------------------------------------------
extern "C" void kernel_launch(void* const* d_in, const int* in_sizes, int n_in,
                              void* d_out, int out_size, void* d_ws, size_t ws_size,
                              hipStream_t stream)
{
  const float* x       = (const float*)d_in[0]; // (64,2048,100)
  const float* embed   = (const float*)d_in[1]; // (50000,1024)
  const float* conv_w  = (const float*)d_in[2]; // (1024,100)
  const float* conv_b  = (const float*)d_in[3]; // (1024)
  const float* lin_w   = (const float*)d_in[4]; // (1024,1024)
  const float* lin_b   = (const float*)d_in[5]; // (1024)
  float* out = (float*)d_out;                   // (64,50000)

  float* pooled = (float*)d_ws;                 // 64*1024 floats
  float* xn     = pooled + (size_t)B_ * D_;     // 64*1024 floats

  k1_conv_elu_maxpool<<<dim3(8, 64), 256, 0, stream>>>(x, conv_w, conv_b, pooled);
  k2_fc_relu_norm<<<64, 256, 0, stream>>>(pooled, lin_w, lin_b, xn);
  k3_logits_wmma<<<3125, 128, 0, stream>>>(xn, embed, out);
  k4_softmax<<<64, 512, 0, stream>>>(out);
}